// Involution_73400991089028
// MI455X (gfx1250) — compile-verified
//
#include <hip/hip_runtime.h>

typedef __attribute__((ext_vector_type(2))) float v2f;
typedef __attribute__((ext_vector_type(4))) float v4f;
typedef __attribute__((ext_vector_type(8))) float v8f;

#define IMG_H 128
#define IMG_W 128
#define CH 64
#define NPIX 131072L   // 8*128*128

// ---------------------------------------------------------------------------
// Kernel 1: fused kernel-generation branch, fully on the matrix pipe.
//   h      = ReLU(BN(x @ w1 + b1))     -> 64x v_wmma_f32_16x16x4_f32 / wave
//   kern16 = h @ w2pad(64x16) + b2     -> 16x v_wmma_f32_16x16x4_f32 / wave
// One wave = 16 pixels. 8 waves / block = 128 pixels.
// ---------------------------------------------------------------------------
__global__ __launch_bounds__(256) void involution_kgen(
    const float* __restrict__ x,  const float* __restrict__ w1,
    const float* __restrict__ b1, const float* __restrict__ gamma,
    const float* __restrict__ beta, const float* __restrict__ bn_mean,
    const float* __restrict__ bn_var, const float* __restrict__ w2,
    const float* __restrict__ b2, float* __restrict__ kern_out)
{
    __shared__ float lw1[64 * 65];     // w1, stride 65: conflict-free B-frag reads
    __shared__ float ls[64];           // folded BN scale
    __shared__ float lt[64];           // folded BN shift (includes b1)
    __shared__ float lw2p[64 * 16];    // w2 zero-padded 64x9 -> 64x16, stride 16
    __shared__ float lb2[16];
    __shared__ float lh[8][16 * 65];   // per-wave h tile, stride 65

    const int tid = threadIdx.x;

    for (int i = tid; i < 64 * 64; i += 256)
        lw1[(i >> 6) * 65 + (i & 63)] = w1[i];
    for (int i = tid; i < 64 * 16; i += 256) {
        const int k = i >> 4, n = i & 15;
        lw2p[i] = (n < 9) ? w2[k * 9 + n] : 0.f;
    }
    if (tid < 64) {
        float s = gamma[tid] * rsqrtf(bn_var[tid] + 1e-3f);
        ls[tid] = s;
        lt[tid] = (b1[tid] - bn_mean[tid]) * s + beta[tid];
    }
    if (tid < 16) lb2[tid] = (tid < 9) ? b2[tid] : 0.f;
    __syncthreads();

    const int  wave   = tid >> 5;
    const int  lane   = tid & 31;
    const int  mrow   = lane & 15;   // A-matrix row (pixel within wave tile)
    const int  khalf  = lane >> 4;   // K half select per ISA VGPR layout
    const int  col    = lane & 15;   // B/C/D column
    const long pixel0 = (long)blockIdx.x * 128 + wave * 16;

    // ---- GEMM 1: x(16x64) @ w1(64x64), 4 N-tiles, K in steps of 4 ----------
    v8f acc0 = {}, acc1 = {}, acc2 = {}, acc3 = {};
    const float* xrow = x + (pixel0 + mrow) * CH;

    #pragma unroll
    for (int ks = 0; ks < 16; ++ks) {
        const int k = ks * 4 + khalf * 2;
        v2f a = *(const v2f*)(xrow + k);               // x[p][k], x[p][k+1]
        v2f bb0, bb1, bb2, bb3;                        // w1[k][n], w1[k+1][n]
        const int r0 = k * 65, r1 = (k + 1) * 65;
        bb0.x = lw1[r0 + col];       bb0.y = lw1[r1 + col];
        bb1.x = lw1[r0 + 16 + col];  bb1.y = lw1[r1 + 16 + col];
        bb2.x = lw1[r0 + 32 + col];  bb2.y = lw1[r1 + 32 + col];
        bb3.x = lw1[r0 + 48 + col];  bb3.y = lw1[r1 + 48 + col];
        acc0 = __builtin_amdgcn_wmma_f32_16x16x4_f32(false, a, false, bb0, (short)0, acc0, false, false);
        acc1 = __builtin_amdgcn_wmma_f32_16x16x4_f32(false, a, false, bb1, (short)0, acc1, false, false);
        acc2 = __builtin_amdgcn_wmma_f32_16x16x4_f32(false, a, false, bb2, (short)0, acc2, false, false);
        acc3 = __builtin_amdgcn_wmma_f32_16x16x4_f32(false, a, false, bb3, (short)0, acc3, false, false);
    }

    // ---- BN + ReLU in registers, spill h tile to LDS -----------------------
    // C/D layout: VGPR r holds row r (lanes 0-15) / row r+8 (lanes 16-31).
    float* hb = &lh[wave][0];
    #pragma unroll
    for (int r = 0; r < 8; ++r) {
        const int row = r + 8 * khalf;
        {   const int c = col;
            hb[row * 65 + c] = fmaxf(acc0[r] * ls[c] + lt[c], 0.f); }
        {   const int c = 16 + col;
            hb[row * 65 + c] = fmaxf(acc1[r] * ls[c] + lt[c], 0.f); }
        {   const int c = 32 + col;
            hb[row * 65 + c] = fmaxf(acc2[r] * ls[c] + lt[c], 0.f); }
        {   const int c = 48 + col;
            hb[row * 65 + c] = fmaxf(acc3[r] * ls[c] + lt[c], 0.f); }
    }
    __syncthreads();

    // ---- GEMM 2: h(16x64) @ w2pad(64x16), also on the matrix pipe ----------
    v8f acck = {};
    const float* harow = &lh[wave][mrow * 65];
    #pragma unroll
    for (int ks = 0; ks < 16; ++ks) {
        const int k = ks * 4 + khalf * 2;
        v2f a;                                          // h[p][k], h[p][k+1]
        a.x = harow[k];
        a.y = harow[k + 1];
        v2f bb;                                         // w2p[k][n], w2p[k+1][n]
        bb.x = lw2p[k * 16 + col];
        bb.y = lw2p[(k + 1) * 16 + col];
        acck = __builtin_amdgcn_wmma_f32_16x16x4_f32(false, a, false, bb, (short)0, acck, false, false);
    }

    // Store kern: lane column n = col; only n < 9 is real output.
    if (col < 9) {
        const float bj = lb2[col];
        #pragma unroll
        for (int r = 0; r < 8; ++r) {
            const int row = r + 8 * khalf;
            kern_out[(pixel0 + row) * 9 + col] = acck[r] + bj;
        }
    }
}

// ---------------------------------------------------------------------------
// Kernel 2: spatially-varying 3x3 multiply-reduce (SAME / zero padding).
//   out[p][c] = sum_{t=0..8} x[nbr_t(p)][c] * kern[p][t]
// 16 threads per pixel, float4 channels each -> fully coalesced 16B accesses.
// ---------------------------------------------------------------------------
__global__ __launch_bounds__(256) void involution_apply(
    const float* __restrict__ x, const float* __restrict__ kern,
    float* __restrict__ out)
{
    const int  tid = threadIdx.x;
    const long pix = (long)blockIdx.x * 16 + (tid >> 4);
    const int  c4  = (tid & 15) * 4;

    const int b  = (int)(pix >> 14);        // H*W = 16384
    const int ij = (int)(pix & 16383);
    const int i  = ij >> 7;                 // W = 128
    const int j  = ij & 127;

    const float* kp = kern + pix * 9;
    const float* xb = x + (((long)b) << 14) * CH;

    v4f acc = {0.f, 0.f, 0.f, 0.f};
    #pragma unroll
    for (int t = 0; t < 9; ++t) {
        const int di = t / 3 - 1;
        const int dj = t % 3 - 1;
        const int ii = i + di;
        const int jj = j + dj;
        if (ii >= 0 && ii < IMG_H && jj >= 0 && jj < IMG_W) {
            v4f xv = *(const v4f*)(xb + ((long)((ii << 7) + jj)) * CH + c4);
            acc += xv * kp[t];
        }
    }
    *(v4f*)(out + pix * CH + c4) = acc;
}

// ---------------------------------------------------------------------------
extern "C" void kernel_launch(void* const* d_in, const int* in_sizes, int n_in,
                              void* d_out, int out_size, void* d_ws, size_t ws_size,
                              hipStream_t stream) {
    const float* x       = (const float*)d_in[0];
    const float* w1      = (const float*)d_in[1];
    const float* b1      = (const float*)d_in[2];
    const float* gamma   = (const float*)d_in[3];
    const float* beta    = (const float*)d_in[4];
    const float* bn_mean = (const float*)d_in[5];
    const float* bn_var  = (const float*)d_in[6];
    const float* w2      = (const float*)d_in[7];
    const float* b2      = (const float*)d_in[8];

    float* out  = (float*)d_out;                 // (B,H,W,C)  = 8,388,608 floats
    float* kern = out + NPIX * CH;               // (B,H,W,9)  = 1,179,648 floats

    // Kernel 1: 128 pixels / block -> 1024 blocks, 8 waves each.
    involution_kgen<<<(int)(NPIX / 128), 256, 0, stream>>>(
        x, w1, b1, gamma, beta, bn_mean, bn_var, w2, b2, kern);

    // Kernel 2: 16 pixels / block -> 8192 blocks.
    involution_apply<<<(int)(NPIX / 16), 256, 0, stream>>>(x, kern, out);
}